// SoftTriple_bias_47811575939500
// MI455X (gfx1250) — compile-verified
//
#include <hip/hip_runtime.h>
#include <math.h>

typedef __attribute__((ext_vector_type(2))) float v2f;
typedef __attribute__((ext_vector_type(4))) float v4f;
typedef __attribute__((ext_vector_type(8))) float v8f;

#define NROWS 16384
#define NDIM  128
#define NK    2048
#define NSUMK 4096

// workspace layout (bytes)
#define NX_OFF  0ul
#define CT_OFF  ((size_t)NROWS * NDIM * 4ul)            // nx: 8 MiB
#define CE_OFF  (CT_OFF + (size_t)NSUMK * NDIM * 4ul)   // centersT: 2 MiB
#define NCE     256                                     // one CE partial per 64-row block
#define TPC     8256                                    // 128*129/2 upper-tri 16x16 tiles per class
#define NREGB   (2*TPC)
#define REG_OFF (CE_OFF + (size_t)NCE * 4ul)

__device__ __forceinline__ float red16(float v) {
  v += __shfl_xor(v, 1, 32);
  v += __shfl_xor(v, 2, 32);
  v += __shfl_xor(v, 4, 32);
  v += __shfl_xor(v, 8, 32);
  return v;
}

// ---------------- normalize x rows: one wave per row ----------------
__global__ void __launch_bounds__(256)
norm_x_kernel(const float* __restrict__ x, float* __restrict__ nx) {
  int row  = (blockIdx.x * blockDim.x + threadIdx.x) >> 5;
  int lane = threadIdx.x & 31;
  const float* rp = x + (size_t)row * NDIM + lane * 4;
  v4f v = *(const v4f*)rp;
  float ss = v.x*v.x + v.y*v.y + v.z*v.z + v.w*v.w;
  ss = red16(ss);
  ss += __shfl_xor(ss, 16, 32);
  float rinv = 1.0f / sqrtf(ss);
  v4f o = { v.x*rinv, v.y*rinv, v.z*rinv, v.w*rinv };
  *(v4f*)(nx + (size_t)row * NDIM + lane * 4) = o;
}

// ------- normalize fc columns AND transpose -> centersT[col][dim] -------
__global__ void __launch_bounds__(256)
norm_fc_kernel(const float* __restrict__ fc, float* __restrict__ ct) {
  int col  = (blockIdx.x * blockDim.x + threadIdx.x) >> 5;
  int lane = threadIdx.x & 31;
  int d0 = lane * 4;
  float v0 = fc[(size_t)(d0 + 0) * NSUMK + col];
  float v1 = fc[(size_t)(d0 + 1) * NSUMK + col];
  float v2 = fc[(size_t)(d0 + 2) * NSUMK + col];
  float v3 = fc[(size_t)(d0 + 3) * NSUMK + col];
  float ss = v0*v0 + v1*v1 + v2*v2 + v3*v3;
  ss = red16(ss);
  ss += __shfl_xor(ss, 16, 32);
  float rinv = 1.0f / sqrtf(ss);
  v4f o = { v0*rinv, v1*rinv, v2*rinv, v3*rinv };
  *(v4f*)(ct + (size_t)col * NDIM + d0) = o;
}

// ---- fused simInd GEMM + per-class softmax stats + logits + CE partial ----
// grid: NROWS/64 blocks of 256 threads (8 waves).
// wave wid: row-tile (wid>>1), class (wid&1). The 4 waves of a class sweep the
// SAME column tiles -> B loads hit WGP$ after the first wave.
// Inner loop: manual half-tile ping-pong so 16 loads are always in flight
// behind 16 WMMAs.
__global__ void __launch_bounds__(256) __attribute__((amdgpu_waves_per_eu(1, 8)))
sim_softmax_ce_kernel(const float* __restrict__ nx, const float* __restrict__ ct,
                      const int* __restrict__ target, float* __restrict__ out,
                      float* __restrict__ ce_part) {
  __shared__ float lsum[8][16][2];
  __shared__ float lce[2];
  int lane = threadIdx.x & 31;
  int wid  = threadIdx.x >> 5;
  int m16  = lane & 15;
  int kh   = lane >> 4;
  int rt   = wid >> 1;      // row tile 0..3
  int cls  = wid & 1;       // class 0/1
  int rowBase = blockIdx.x * 64 + rt * 16;

  // A fragments for a 16x128 tile of nx, kept in registers (64 VGPRs)
  const float* arow = nx + (size_t)(rowBase + m16) * NDIM + 2 * kh;
  v2f a[32];
#pragma unroll
  for (int kk = 0; kk < 32; ++kk) a[kk] = *(const v2f*)(arow + kk * 4);

  const float* tbase = ct + (size_t)cls * NK * NDIM + (size_t)m16 * NDIM + 2 * kh;
  float s1[8], s2[8];
#pragma unroll
  for (int j = 0; j < 8; ++j) { s1[j] = 0.f; s2[j] = 0.f; }

  v2f b0[16], b1[16];
#pragma unroll
  for (int q = 0; q < 16; ++q) b0[q] = *(const v2f*)(tbase + q * 4);  // t=0, K 0..63

  for (int t = 0; t < 128; ++t) {
    const float* pc = tbase + (size_t)t * 16 * NDIM;
    // issue loads for second K-half of this tile
#pragma unroll
    for (int q = 0; q < 16; ++q) b1[q] = *(const v2f*)(pc + 64 + q * 4);
    v8f acc = {};
#pragma unroll
    for (int q = 0; q < 16; ++q) {
      acc = __builtin_amdgcn_wmma_f32_16x16x4_f32(false, a[q], false, b0[q],
                                                  (short)0, acc, false, false);
    }
    // issue loads for first K-half of the NEXT tile while b1-WMMAs run
    if (t < 127) {
      const float* pn = pc + 16 * NDIM;
#pragma unroll
      for (int q = 0; q < 16; ++q) b0[q] = *(const v2f*)(pn + q * 4);
    }
#pragma unroll
    for (int q = 0; q < 16; ++q) {
      acc = __builtin_amdgcn_wmma_f32_16x16x4_f32(false, a[16 + q], false, b1[q],
                                                  (short)0, acc, false, false);
    }
#pragma unroll
    for (int j = 0; j < 8; ++j) {
      float s = acc[j];
      float e = __expf(10.0f * s - 10.0f);   // exp(gamma_inv*(sim-1)), shift-invariant
      s1[j] += e;
      s2[j] += e * s;
    }
  }

  // reduce over the 16 column-lanes of each half-wave
#pragma unroll
  for (int j = 0; j < 8; ++j) { s1[j] = red16(s1[j]); s2[j] = red16(s2[j]); }
  if (m16 == 0) {
#pragma unroll
    for (int j = 0; j < 8; ++j) {
      int m = j + 8 * kh;
      lsum[wid][m][0] = s1[j];
      lsum[wid][m][1] = s2[j];
    }
  }
  __syncthreads();

  float ce = 0.f;
  if (threadIdx.x < 64) {          // tid -> (row tile, row)
    int trt = threadIdx.x >> 4;
    int m   = threadIdx.x & 15;
    int r   = blockIdx.x * 64 + trt * 16 + m;
    float S10 = lsum[trt * 2 + 0][m][0];
    float S20 = lsum[trt * 2 + 0][m][1];
    float S11 = lsum[trt * 2 + 1][m][0];
    float S21 = lsum[trt * 2 + 1][m][1];
    float l0 = 20.0f * (S20 / S10 - 0.01f);
    float l1 = 20.0f * (S21 / S11 - 0.01f);
    out[1 + 2 * (size_t)r + 0] = l0;
    out[1 + 2 * (size_t)r + 1] = l1;
    float mx  = fmaxf(l0, l1);
    float lse = mx + __logf(__expf(l0 - mx) + __expf(l1 - mx));
    int tg = target[r];
    ce = lse - (tg == 0 ? l0 : l1);
  }
  if (wid < 2) {                   // waves 0/1 fully active -> safe shuffles
    ce = red16(ce);
    ce += __shfl_xor(ce, 16, 32);
    if (lane == 0) lce[wid] = ce;
  }
  __syncthreads();
  if (threadIdx.x == 0) ce_part[blockIdx.x] = lce[0] + lce[1];
}

// ---- center-pair regularizer: one wave per upper-triangular 16x16 tile ----
__device__ __forceinline__ int tri_start(int i) { return i * 128 - (i * (i - 1)) / 2; }

__global__ void __launch_bounds__(32) __attribute__((amdgpu_waves_per_eu(1, 8)))
center_reg_kernel(const float* __restrict__ ct, float* __restrict__ reg_part) {
  int bid = blockIdx.x;
  int c = bid / TPC;
  int t = bid - c * TPC;
  // decode triangular tile index -> (i, j), j >= i, T = 128
  double disc = (257.0 * 257.0) - 8.0 * (double)t;
  int i = (int)((257.0 - sqrt(disc)) * 0.5);
  if (i < 0) i = 0;
  if (i > 127) i = 127;
  while (i < 127 && tri_start(i + 1) <= t) ++i;
  while (i > 0 && tri_start(i) > t) --i;
  int j = i + (t - tri_start(i));

  int lane = threadIdx.x;
  int m16 = lane & 15;
  int kh  = lane >> 4;
  const float* arow = ct + (size_t)(c * NK + i * 16 + m16) * NDIM + 2 * kh;
  const float* brow = ct + (size_t)(c * NK + j * 16 + m16) * NDIM + 2 * kh;

  // batch both operand tiles -> one big load clause, then WMMA chain
  v2f a[32], b[32];
#pragma unroll
  for (int kk = 0; kk < 32; ++kk) a[kk] = *(const v2f*)(arow + kk * 4);
#pragma unroll
  for (int kk = 0; kk < 32; ++kk) b[kk] = *(const v2f*)(brow + kk * 4);

  v8f acc = {};
#pragma unroll
  for (int kk = 0; kk < 32; ++kk) {
    acc = __builtin_amdgcn_wmma_f32_16x16x4_f32(false, a[kk], false, b[kk],
                                                (short)0, acc, false, false);
  }

  float sum = 0.f;
#pragma unroll
  for (int jv = 0; jv < 8; ++jv) {
    int gi = i * 16 + jv + 8 * kh;   // D row (A side)
    int gj = j * 16 + m16;           // D col (B side)
    if (gi < gj) {
      float s = acc[jv];
      sum += sqrtf(fmaxf(2.0f + 1e-5f - 2.0f * s, 0.0f));
    }
  }
  sum = red16(sum);
  sum += __shfl_xor(sum, 16, 32);
  if (lane == 0) reg_part[bid] = sum;
}

// ---- deterministic final reduction ----
__global__ void __launch_bounds__(256)
finalize_kernel(const float* __restrict__ ce_part, const float* __restrict__ reg_part,
                float* __restrict__ out) {
  __shared__ float sh[256];
  float s = 0.f;
  for (int i = threadIdx.x; i < NCE; i += 256) s += ce_part[i];
  sh[threadIdx.x] = s;
  __syncthreads();
  for (int off = 128; off > 0; off >>= 1) {
    if (threadIdx.x < (unsigned)off) sh[threadIdx.x] += sh[threadIdx.x + off];
    __syncthreads();
  }
  float ce_sum = sh[0];
  __syncthreads();
  float r = 0.f;
  for (int i = threadIdx.x; i < NREGB; i += 256) r += reg_part[i];
  sh[threadIdx.x] = r;
  __syncthreads();
  for (int off = 128; off > 0; off >>= 1) {
    if (threadIdx.x < (unsigned)off) sh[threadIdx.x] += sh[threadIdx.x + off];
    __syncthreads();
  }
  if (threadIdx.x == 0) {
    float reg_sum = sh[0];
    // N_PAIRS = 2*2048*2047/2 = 4192256 ; divisor = N_PAIRS*2
    out[0] = ce_sum / 16384.0f + 0.2f * (reg_sum / 8384512.0f);
  }
}

extern "C" void kernel_launch(void* const* d_in, const int* in_sizes, int n_in,
                              void* d_out, int out_size, void* d_ws, size_t ws_size,
                              hipStream_t stream) {
  const float* x      = (const float*)d_in[0];
  const int*   target = (const int*)d_in[1];   // int inputs arrive as int32
  const float* fc     = (const float*)d_in[2];
  float* out = (float*)d_out;

  char* ws = (char*)d_ws;
  float* nx       = (float*)(ws + NX_OFF);
  float* ct       = (float*)(ws + CT_OFF);
  float* ce_part  = (float*)(ws + CE_OFF);
  float* reg_part = (float*)(ws + REG_OFF);

  norm_x_kernel  <<<NROWS / 8, 256, 0, stream>>>(x, nx);
  norm_fc_kernel <<<NSUMK / 8, 256, 0, stream>>>(fc, ct);
  sim_softmax_ce_kernel<<<NROWS / 64, 256, 0, stream>>>(nx, ct, target, out, ce_part);
  center_reg_kernel    <<<NREGB, 32, 0, stream>>>(ct, reg_part);
  finalize_kernel      <<<1, 256, 0, stream>>>(ce_part, reg_part, out);
}